// IclLoss_16638703304761
// MI455X (gfx1250) — compile-verified
//
#include <hip/hip_runtime.h>
#include <math.h>

#define TAU_INV    20.0f         // 1/0.05, exact in fp32
#define ALPHA_F    0.5f
#define EPS_F      1e-12f
#define B_LINKS    4096
#define NROWS      8192          // 2*B
#define DIM        128
#define NTILES     (NROWS / 16)  // 512 column tiles of 16

typedef __attribute__((ext_vector_type(2))) float v2f;
typedef __attribute__((ext_vector_type(8))) float v8f;

// ---------------------------------------------------------------------------
// Kernel 1: gather rows by link index, L2-normalize, write Zall (8192 x 128).
// One wave (32 lanes) per output row; each lane handles 4 contiguous floats.
// ---------------------------------------------------------------------------
__global__ void k_gather_normalize(const float* __restrict__ emb,
                                   const long long* __restrict__ links,
                                   float* __restrict__ zall) {
    const int wave = threadIdx.x >> 5;
    const int lane = threadIdx.x & 31;
    const int row  = blockIdx.x * 8 + wave;          // 0..8191
    if (row >= NROWS) return;

    long long idx = (row < B_LINKS) ? links[2 * row]
                                    : links[2 * (row - B_LINKS) + 1];

    const float4* src = (const float4*)(emb + (size_t)idx * DIM);
    float4 v = src[lane];

    float ss = v.x * v.x + v.y * v.y + v.z * v.z + v.w * v.w;
    ss += __shfl_xor(ss, 1);
    ss += __shfl_xor(ss, 2);
    ss += __shfl_xor(ss, 4);
    ss += __shfl_xor(ss, 8);
    ss += __shfl_xor(ss, 16);

    const float scale = 1.0f / fmaxf(sqrtf(ss), EPS_F);
    float4 o;
    o.x = v.x * scale; o.y = v.y * scale; o.z = v.z * scale; o.w = v.w * scale;
    ((float4*)(zall + (size_t)row * DIM))[lane] = o;
}

// ---------------------------------------------------------------------------
// Kernel 2: streaming Gram + log-sum-exp with a FIXED shift.
// Rows are unit vectors => every logit = cos/tau is in [-20, 20], so shifting
// by the constant 20 is a valid softmax max: exp(v-20) in [e^-40, 1].
// Each lane privately accumulates sum(exp(v-20)) per row; one cross-lane
// reduction at the end.  lse = 20 + log(sum).
//
// Workgroup = 16 rows (one M tile), 8 waves split the 512 N tiles.
// K=128 as a chain of 32 x V_WMMA_F32_16X16X4_F32 (fp32 in/out, matches ref),
// software-pipelined: B-fragment group kq+1 is loaded before group kq's WMMAs
// issue, so each load sits a full WMMA group ahead of its consuming wait.
// fp32 16x16x4 fragment per-lane addressing (ISA 7.12.2), identical for A and
// B of a Gram matrix:  Z[base + (lane&15)][4*kb + 2*(lane>>4) + {0,1}].
// Accumulator: VGPR p, lanes 0-15 -> row p, lanes 16-31 -> row p+8; N = lane.
// ---------------------------------------------------------------------------
__global__ void k_gram_lse(const float* __restrict__ zall,
                           float* __restrict__ contrib) {
    __shared__ float lsum[8][16];
    __shared__ float flse[16];

    const int wave = threadIdx.x >> 5;
    const int lane = threadIdx.x & 31;
    const int h    = lane >> 4;     // half-wave
    const int l16  = lane & 15;
    const int rowbase = blockIdx.x * 16;

    // A fragments for the 16 rows of this block (live across the whole loop).
    const float* aptr = zall + (size_t)(rowbase + l16) * DIM + 2 * h;
    v2f afrag[32];
#pragma unroll
    for (int kb = 0; kb < 32; ++kb)
        afrag[kb] = *(const v2f*)(aptr + 4 * kb);

    float s[8];
#pragma unroll
    for (int p = 0; p < 8; ++p) s[p] = 0.0f;

    for (int ct = wave; ct < NTILES; ct += 8) {
        const int colbase = ct * 16;
        const float* bptr = zall + (size_t)(colbase + l16) * DIM + 2 * h;

        // Prefetch the wave's next column tile: 2 prefetches/lane cover all
        // four 128B lines of each of the 16 next-tile rows (512B per row).
        if (ct + 8 < NTILES) {
            const float* prow =
                zall + (size_t)(colbase + 128 + l16) * DIM + h * 32;
            __builtin_prefetch(prow, 0, 3);
            __builtin_prefetch(prow + 64, 0, 3);
        }

        v8f acc = {};
        v2f cur0 = *(const v2f*)(bptr + 0);
        v2f cur1 = *(const v2f*)(bptr + 4);
        v2f cur2 = *(const v2f*)(bptr + 8);
        v2f cur3 = *(const v2f*)(bptr + 12);
#pragma unroll
        for (int kq = 0; kq < 8; ++kq) {
            v2f nxt0 = cur0, nxt1 = cur1, nxt2 = cur2, nxt3 = cur3;
            if (kq < 7) {   // compile-time resolved under full unroll
                const float* np = bptr + 16 * (kq + 1);
                nxt0 = *(const v2f*)(np + 0);
                nxt1 = *(const v2f*)(np + 4);
                nxt2 = *(const v2f*)(np + 8);
                nxt3 = *(const v2f*)(np + 12);
            }
            acc = __builtin_amdgcn_wmma_f32_16x16x4_f32(
                false, afrag[4 * kq + 0], false, cur0, (short)0, acc, false, false);
            acc = __builtin_amdgcn_wmma_f32_16x16x4_f32(
                false, afrag[4 * kq + 1], false, cur1, (short)0, acc, false, false);
            acc = __builtin_amdgcn_wmma_f32_16x16x4_f32(
                false, afrag[4 * kq + 2], false, cur2, (short)0, acc, false, false);
            acc = __builtin_amdgcn_wmma_f32_16x16x4_f32(
                false, afrag[4 * kq + 3], false, cur3, (short)0, acc, false, false);
            cur0 = nxt0; cur1 = nxt1; cur2 = nxt2; cur3 = nxt3;
        }

        // Epilogue: accumulate exp(logit - 20) privately per lane.
        const int col = colbase + l16;
#pragma unroll
        for (int p = 0; p < 8; ++p) {
            const int row = rowbase + p + 8 * h;
            float e = __expf(fmaf(acc[p], TAU_INV, -TAU_INV));
            if (col == row) e = 0.0f;   // self column: exp underflows to 0 in ref
            s[p] += e;
        }
    }

    // One cross-lane reduction per row (16 lanes share a row).
#pragma unroll
    for (int p = 0; p < 8; ++p) {
        s[p] += __shfl_xor(s[p], 1);
        s[p] += __shfl_xor(s[p], 2);
        s[p] += __shfl_xor(s[p], 4);
        s[p] += __shfl_xor(s[p], 8);
    }
    if (l16 == 0) {
#pragma unroll
        for (int p = 0; p < 8; ++p) lsum[wave][p + 8 * h] = s[p];
    }
    __syncthreads();

    if (threadIdx.x < 16) {
        const int rt = threadIdx.x;
        float S = 0.0f;
#pragma unroll
        for (int w2 = 0; w2 < 8; ++w2) S += lsum[w2][rt];
        flse[rt] = TAU_INV + __logf(S);
    }
    __syncthreads();

    // Positive logit: dot(Z[r], Z[r^4096]) — one wave per row (2 rows/wave).
    for (int rt = wave; rt < 16; rt += 8) {
        const int r  = rowbase + rt;
        const int pr = r ^ B_LINKS;
        float4 x = ((const float4*)(zall + (size_t)r  * DIM))[lane];
        float4 y = ((const float4*)(zall + (size_t)pr * DIM))[lane];
        float d = x.x * y.x + x.y * y.y + x.z * y.z + x.w * y.w;
        d += __shfl_xor(d, 1);
        d += __shfl_xor(d, 2);
        d += __shfl_xor(d, 4);
        d += __shfl_xor(d, 8);
        d += __shfl_xor(d, 16);
        if (lane == 0) {
            const float wgt =
                ((r < B_LINKS) ? ALPHA_F : (1.0f - ALPHA_F)) / (float)B_LINKS;
            contrib[r] = wgt * (flse[rt] - d * TAU_INV);
        }
    }
}

// ---------------------------------------------------------------------------
// Kernel 3: deterministic fixed-order reduction of 8192 contributions.
// ---------------------------------------------------------------------------
__global__ void k_final_reduce(const float* __restrict__ contrib,
                               float* __restrict__ out) {
    __shared__ float sm[256];
    float acc = 0.0f;
    for (int i = threadIdx.x; i < NROWS; i += 256) acc += contrib[i];
    sm[threadIdx.x] = acc;
    __syncthreads();
    for (int off = 128; off > 0; off >>= 1) {
        if ((int)threadIdx.x < off) sm[threadIdx.x] += sm[threadIdx.x + off];
        __syncthreads();
    }
    if (threadIdx.x == 0) out[0] = sm[0];
}

// ---------------------------------------------------------------------------
extern "C" void kernel_launch(void* const* d_in, const int* in_sizes, int n_in,
                              void* d_out, int out_size, void* d_ws,
                              size_t ws_size, hipStream_t stream) {
    const float* emb       = (const float*)d_in[0];       // 200000 x 128 f32
    const long long* links = (const long long*)d_in[1];   // 4096 x 2 i64
    float* out = (float*)d_out;                           // scalar f32

    float* zall    = (float*)d_ws;                        // 8192 x 128 f32
    float* contrib = zall + (size_t)NROWS * DIM;          // 8192 f32

    k_gather_normalize<<<NROWS / 8, 256, 0, stream>>>(emb, links, zall);
    k_gram_lse<<<NTILES, 256, 0, stream>>>(zall, contrib);
    k_final_reduce<<<1, 256, 0, stream>>>(contrib, out);
}